// LogEigLayer_26018911879185
// MI455X (gfx1250) — compile-verified
//
#include <hip/hip_runtime.h>
#include <cmath>

// ---------------------------------------------------------------------------
// logm(M) for batched SPD 64x64 fp32 matrices via inverse scaling-and-squaring
// (coupled Newton-Schulz sqrt x4 + 8-term Mercator/Horner series).
// All GEMMs: v_wmma_f32_16x16x4_f32 (fp32 WMMA), one workgroup (4 x wave32)
// per matrix, matrices resident in LDS (4 buffers of 64x68 f32 ~ 70KB/WG ->
// 4 workgroups per WGP under the 320KB budget).
//
// Every operand matrix here is symmetric (polynomials of symmetric A), so the
// B fragment is read TRANSPOSED: both components land 2 floats apart in one
// LDS row -> single ds_load_2addr_b32 per fragment into an adjacent VGPR
// pair, no packing moves. k-outer loop with 4 independent accumulators
// breaks the WMMA RAW chain; alpha/diag epilogue fused into the store.
// ---------------------------------------------------------------------------

typedef __attribute__((ext_vector_type(2))) float v2f;
typedef __attribute__((ext_vector_type(4))) float v4f;
typedef __attribute__((ext_vector_type(8))) float v8f;

#define N64 64
#define LD  68                 // padded LDS row stride (floats), avoids bank conflicts
#define MAT_ELEMS (N64 * LD)   // 4352 floats per buffer
#define NBUF 4

// ---- WMMA fragment helpers (layouts per CDNA5 ISA 7.12.2, wave32) ----------

// A fragment: 16x4 f32.  Lanes 0-15: row=lane,    V0=K0, V1=K1.
//                        Lanes 16-31: row=lane-16, V0=K2, V1=K3.
__device__ __forceinline__ v2f ldA(const float* M, int m0, int k0, int lane) {
    int row = m0 + (lane & 15);
    int kb  = k0 + ((lane >> 4) << 1);
    v2f a;
    a.x = M[row * LD + kb];
    a.y = M[row * LD + kb + 1];
    return a;
}

// B fragment: 4x16 f32, rows striped across lanes within a VGPR.
// V0: lanes 0-15 = row K0, lanes 16-31 = row K1.
// V1: lanes 0-15 = row K2, lanes 16-31 = row K3.
// Operand matrices are symmetric: B[k][n] == B[n][k], so read the transposed
// element -> both components 2 floats apart in the same LDS row.
__device__ __forceinline__ v2f ldB_sym(const float* M, int k0, int n0, int lane) {
    int row = n0 + (lane & 15);           // = column index n of the fragment
    int kb  = k0 + (lane >> 4);
    v2f b;
    b.x = M[row * LD + kb];               // B[kb  ][n] via symmetry
    b.y = M[row * LD + kb + 2];           // B[kb+2][n] via symmetry
    return b;
}

// C/D 16x16 f32: VGPR r -> lanes 0-15: (M=r, N=lane); lanes 16-31: (M=r+8, N=lane-16).
// Fused epilogue: D = alpha*acc + diag*I.
__device__ __forceinline__ void stC(float* D, int m0, int n0, int lane, v8f c,
                                    float alpha, float diag) {
    int col   = n0 + (lane & 15);
    int rbase = m0 + ((lane >> 4) << 3);
#pragma unroll
    for (int r = 0; r < 8; ++r) {
        int row = rbase + r;
        float v = alpha * c[r];
        if (row == col) v += diag;
        D[row * LD + col] = v;
    }
}

// ---- Whole-workgroup 64x64 GEMM: D = alpha*(A*B) + diag*I ------------------
// Entry barrier only (covers visibility of all prior LDS writes AND completes
// all prior reads of the buffer D overwrites). D must not alias A or B.
// A and B must be symmetric (true for all operands in this algorithm).
__device__ __forceinline__ void wg_mm(float* __restrict__ D,
                                      const float* __restrict__ A,
                                      const float* __restrict__ Bm,
                                      int lane, int wave,
                                      float alpha, float diag) {
    __syncthreads();
    const int m0 = wave << 4;
    v8f acc0 = {}, acc1 = {}, acc2 = {}, acc3 = {};
#pragma unroll
    for (int k0 = 0; k0 < 64; k0 += 4) {
        v2f a  = ldA(A, m0, k0, lane);             // reused by 4 column tiles
        v2f b0 = ldB_sym(Bm, k0, 0,  lane);
        v2f b1 = ldB_sym(Bm, k0, 16, lane);
        v2f b2 = ldB_sym(Bm, k0, 32, lane);
        v2f b3 = ldB_sym(Bm, k0, 48, lane);
        // 4 independent accumulators -> no WMMA->WMMA RAW chain
        acc0 = __builtin_amdgcn_wmma_f32_16x16x4_f32(false, a, false, b0, (short)0, acc0, false, false);
        acc1 = __builtin_amdgcn_wmma_f32_16x16x4_f32(false, a, false, b1, (short)0, acc1, false, false);
        acc2 = __builtin_amdgcn_wmma_f32_16x16x4_f32(false, a, false, b2, (short)0, acc2, false, false);
        acc3 = __builtin_amdgcn_wmma_f32_16x16x4_f32(false, a, false, b3, (short)0, acc3, false, false);
    }
    stC(D, m0, 0,  lane, acc0, alpha, diag);
    stC(D, m0, 16, lane, acc1, alpha, diag);
    stC(D, m0, 32, lane, acc2, alpha, diag);
    stC(D, m0, 48, lane, acc3, alpha, diag);
}

__global__ __launch_bounds__(128)
void logm_spd64_kernel(const float* __restrict__ in, float* __restrict__ out) {
    __shared__ float smem[NBUF * MAT_ELEMS + 72];

    const int bid  = blockIdx.x;
    const int tid  = threadIdx.x;
    const int lane = tid & 31;
    const int wave = tid >> 5;

    float* Y = smem;                  // current "A" / Newton-Schulz Y
    float* Z = smem + 1 * MAT_ELEMS;  // Newton-Schulz Z
    float* T = smem + 2 * MAT_ELEMS;  // scratch (holds W)
    float* S = smem + 3 * MAT_ELEMS;  // spare (rotation)
    float* red = smem + NBUF * MAT_ELEMS;

    // Load M into Y: float4 global loads -> ds_store_b128 (row stride 68 keeps
    // every 4-aligned column 16B-aligned; a float4 never crosses a row).
    const v4f* src4 = (const v4f*)(in + (size_t)bid * 4096);
    for (int e4 = tid; e4 < 1024; e4 += 128) {
        int r  = e4 >> 4;             // 16 float4 per row
        int c4 = (e4 & 15) << 2;
        *(v4f*)&Y[r * LD + c4] = src4[e4];
    }
    __syncthreads();

    // Gershgorin bound on the spectral norm: s = max_i sum_j |M_ij|
    if (tid < 64) {
        float srow = 0.f;
        for (int c = 0; c < 64; ++c) srow += fabsf(Y[tid * LD + c]);
        red[tid] = srow;
    }
    __syncthreads();
    if (tid == 0) {
        float m = red[0];
        for (int i = 1; i < 64; ++i) m = fmaxf(m, red[i]);
        red[64] = m;
    }
    __syncthreads();
    const float s     = red[64];
    const float inv_s = 1.0f / s;

    // Y = M/s (eigenvalues in (0,1]);  Z = I
    for (int e4 = tid; e4 < 1024; e4 += 128) {
        int r  = e4 >> 4;
        int c4 = (e4 & 15) << 2;
        v4f y = *(const v4f*)&Y[r * LD + c4];
        y *= inv_s;
        *(v4f*)&Y[r * LD + c4] = y;
        v4f z = {0.f, 0.f, 0.f, 0.f};
        if ((r >> 2) == (e4 & 15)) z[r & 3] = 1.f;
        *(v4f*)&Z[r * LD + c4] = z;
    }

    // ---- k=4 repeated square roots via coupled Newton-Schulz ---------------
    // Y_{n+1} = Y_n * W,  Z_{n+1} = W * Z_n,  W = 1.5 I - 0.5 Z_n Y_n
    const int K_SQRT   = 4;
    const int iters[4] = {9, 7, 5, 4};
#pragma unroll 1
    for (int lvl = 0; lvl < K_SQRT; ++lvl) {
        if (lvl > 0) {
            __syncthreads();                       // Z buffer free of readers/writers
            for (int e4 = tid; e4 < 1024; e4 += 128) {
                int r  = e4 >> 4;
                int c4 = (e4 & 15) << 2;
                v4f z = {0.f, 0.f, 0.f, 0.f};
                if ((r >> 2) == (e4 & 15)) z[r & 3] = 1.f;
                *(v4f*)&Z[r * LD + c4] = z;
            }
        }
#pragma unroll 1
        for (int it = 0; it < iters[lvl]; ++it) {
            wg_mm(T, Z, Y, lane, wave, -0.5f, 1.5f);  // T = W = 1.5I - 0.5*Z*Y
            wg_mm(S, Y, T, lane, wave, 1.0f, 0.0f);   // S = Y*W      (new Y)
            wg_mm(Y, T, Z, lane, wave, 1.0f, 0.0f);   // Yold <- W*Z  (new Z)
            float* tmp = Z; Z = Y; Y = S; S = tmp;    // rotate buffers
        }
    }

    // ---- Mercator series: log(Y) = -sum_{j=1..m} X^j / j,  X = I - Y -------
    const int M_TERMS = 8;
    __syncthreads();                               // last GEMM done; Y stable
    float* X = Z;                                  // Z (inverse root) dead
    float* P = Y;                                  // Y reusable once X is formed
    float* Q = S;
    for (int e4 = tid; e4 < 1024; e4 += 128) {
        int r  = e4 >> 4;
        int c4 = (e4 & 15) << 2;
        v4f y = *(const v4f*)&Y[r * LD + c4];      // read then overwrite: same thread
        v4f x = -y;
        v4f p = {0.f, 0.f, 0.f, 0.f};
        if ((r >> 2) == (e4 & 15)) {
            x[r & 3] += 1.f;                       // X = I - Y
            p[r & 3] = 1.0f / (float)M_TERMS;      // P = (1/m) I
        }
        *(v4f*)&X[r * LD + c4] = x;
        *(v4f*)&P[r * LD + c4] = p;
    }
#pragma unroll 1
    for (int j = M_TERMS - 1; j >= 1; --j) {       // Horner: P' = X*P + (1/j) I
        wg_mm(Q, X, P, lane, wave, 1.0f, 1.0f / (float)j);
        float* tmp = P; P = Q; Q = tmp;
    }
    wg_mm(Q, X, P, lane, wave, 1.0f, 0.0f);        // S = X*P ;  log(Y) = -S

    // ---- out = -2^k * S + log(s) * I ---------------------------------------
    __syncthreads();
    const float scale = -(float)(1 << K_SQRT);
    const float logs  = logf(s);
    v4f* dst4 = (v4f*)out + (size_t)bid * 1024;
    for (int e4 = tid; e4 < 1024; e4 += 128) {
        int r  = e4 >> 4;
        int c4 = (e4 & 15) << 2;
        v4f v = *(const v4f*)&Q[r * LD + c4];
        v *= scale;
        if ((r >> 2) == (e4 & 15)) v[r & 3] += logs;
        dst4[e4] = v;
    }
}

extern "C" void kernel_launch(void* const* d_in, const int* in_sizes, int n_in,
                              void* d_out, int out_size, void* d_ws, size_t ws_size,
                              hipStream_t stream) {
    const float* in  = (const float*)d_in[0];
    float*       out = (float*)d_out;
    const int nmat = in_sizes[0] / 4096;           // B = 8192 matrices of 64x64
    logm_spd64_kernel<<<nmat, 128, 0, stream>>>(in, out);
}